// Method_GCN_62483184222789
// MI455X (gfx1250) — compile-verified
//
#include <hip/hip_runtime.h>
#include <cstdint>
#include <cstddef>

#define N_NODES 100000
#define N_EDGES 3200000
#define F_IN    512
#define F_HID   128
#define F_OUT   32

typedef float v2f __attribute__((ext_vector_type(2)));
typedef float v4f __attribute__((ext_vector_type(4)));
typedef float v8f __attribute__((ext_vector_type(8)));

// ---------------------------------------------------------------------------
// One-time weight transpose: Wt[N,K] = W[K,N]^T  (so the WMMA B-operand's two
// K-adjacent values per lane are contiguous -> single global_load_b64).
// ---------------------------------------------------------------------------
template <int K, int N>
__global__ void transposeW(const float* __restrict__ W, float* __restrict__ Wt) {
    const int tid = (int)blockIdx.x * (int)blockDim.x + (int)threadIdx.x;
    if (tid >= K * N) return;
    const int k = tid / N, n = tid % N;
    Wt[(size_t)n * K + k] = W[tid];
}

// ---------------------------------------------------------------------------
// Dense GEMM Y[M,N] = act(X)[M,K] @ W[K,N] using V_WMMA_F32_16X16X4_F32.
// W supplied transposed as Wt[N,K].  One wave computes a 16 x N stripe.
// Per K-step: ALL loads (A + NT B fragments) are issued before ANY WMMA, so
// all fragments are simultaneously live -> distinct dest VGPRs -> the loads
// clause and the WMMAs get staggered s_wait_loadcnt.  No values are carried
// across iterations, so no spill pressure; the compiler pipelines iterations.
// If BIAS_RELU_A: A-operand is relu(X + biasA[k]) fused into the load.
// A layout (16x4 f32): lane L -> row M=L%16, K = 2*(L/16)+{0,1}  (v2f)
// B layout (4x16 f32): lane L -> col N=L%16, K = 2*(L/16)+{0,1}  (v2f)
// C/D (16x16, v8f):    lane L -> col L%16, VGPR r -> row r + 8*(L/16)
// ---------------------------------------------------------------------------
template <int K, int N, bool BIAS_RELU_A>
__global__ void wmma_gemm_f32(const float* __restrict__ X,
                              const float* __restrict__ Wt,
                              const float* __restrict__ biasA,
                              float* __restrict__ Y, int M) {
    constexpr int NT = N / 16;
    const int wave = (int)blockIdx.x * (int)(blockDim.x >> 5) + (int)(threadIdx.x >> 5);
    const int lane = (int)(threadIdx.x & 31u);
    const int m0 = wave * 16;
    if (m0 >= M) return;                 // whole-wave uniform: EXEC stays all-1s

    const int l16 = lane & 15;           // A row / B col within tile
    const int kh  = (lane >> 4) << 1;    // 0 or 2 (K sub-offset per half-wave)

    const float* xrow  = X  + (size_t)(m0 + l16) * K;
    const float* wtrow = Wt + (size_t)l16 * K;   // nt adds constant 16*K offsets

    v8f acc[NT] = {};

    for (int k = 0; k < K; k += 4) {
        // ---- load phase: everything for this K-step, no consumer yet ----
        v2f a = *(const v2f*)(xrow + k + kh);
        v2f b[NT];
#pragma unroll
        for (int nt = 0; nt < NT; ++nt)
            b[nt] = *(const v2f*)(wtrow + (size_t)nt * 16 * K + k + kh);

        if (BIAS_RELU_A) {
            const v2f bb = *(const v2f*)(biasA + k + kh);
            a.x = fmaxf(a.x + bb.x, 0.f);
            a.y = fmaxf(a.y + bb.y, 0.f);
        }

        // ---- compute phase: NT WMMAs ----
#pragma unroll
        for (int nt = 0; nt < NT; ++nt)
            acc[nt] = __builtin_amdgcn_wmma_f32_16x16x4_f32(
                false, a, false, b[nt], (short)0, acc[nt], false, false);
    }

    const int mbase = m0 + (lane >> 4) * 8;
#pragma unroll
    for (int nt = 0; nt < NT; ++nt) {
#pragma unroll
        for (int r = 0; r < 8; ++r) {
            Y[(size_t)(mbase + r) * N + nt * 16 + l16] = acc[nt][r];
        }
    }
}

// ---------------------------------------------------------------------------
// COO SpMM scatter: out[row[e], :] += val[e] * dense[col[e], :]
// F/4 threads per edge; each thread moves a float4 (b128 gather, L2-resident
// dense matrix) and issues 4 native global_atomic_add_f32.
// ---------------------------------------------------------------------------
template <int F>
__global__ void spmm_scatter(const int* __restrict__ rows,
                             const int* __restrict__ cols,
                             const float* __restrict__ vals,
                             const float* __restrict__ dense,
                             float* __restrict__ out) {
    constexpr int CH = F / 4;
    const long long tid = (long long)blockIdx.x * blockDim.x + threadIdx.x;
    const long long total = (long long)N_EDGES * CH;
    if (tid >= total) return;
    const int e = (int)(tid / CH);
    const int c = (int)(tid % CH) * 4;
    const int r  = rows[e];
    const int cl = cols[e];
    const float v = vals[e];
    const v4f d = *(const v4f*)(dense + (size_t)cl * F + c);
    float* dst = out + (size_t)r * F + c;
    unsafeAtomicAdd(dst + 0, v * d.x);
    unsafeAtomicAdd(dst + 1, v * d.y);
    unsafeAtomicAdd(dst + 2, v * d.z);
    unsafeAtomicAdd(dst + 3, v * d.w);
}

// out[n, :] = log_softmax(agg[n, :] + b) over 32 features; one wave per node.
__global__ void bias_logsoftmax32(const float* __restrict__ agg,
                                  const float* __restrict__ b,
                                  float* __restrict__ out) {
    const int wave = (int)blockIdx.x * (int)(blockDim.x >> 5) + (int)(threadIdx.x >> 5);
    const int lane = (int)(threadIdx.x & 31u);
    if (wave >= N_NODES) return;
    const size_t idx = (size_t)wave * 32 + lane;
    const float v = agg[idx] + b[lane];

    float m = v;
#pragma unroll
    for (int off = 16; off > 0; off >>= 1)
        m = fmaxf(m, __shfl_xor(m, off, 32));

    float s = expf(v - m);
#pragma unroll
    for (int off = 16; off > 0; off >>= 1)
        s += __shfl_xor(s, off, 32);

    out[idx] = v - m - logf(s);
}

// ---------------------------------------------------------------------------
extern "C" void kernel_launch(void* const* d_in, const int* in_sizes, int n_in,
                              void* d_out, int out_size, void* d_ws, size_t ws_size,
                              hipStream_t stream) {
    (void)in_sizes; (void)n_in; (void)out_size; (void)ws_size;

    const float* x    = (const float*)d_in[0];
    const int*   erow = (const int*)  d_in[1];
    const int*   ecol = (const int*)  d_in[2];
    const float* ev   = (const float*)d_in[3];
    const float* W1   = (const float*)d_in[4];
    const float* b1   = (const float*)d_in[5];
    const float* W2   = (const float*)d_in[6];
    const float* b2   = (const float*)d_in[7];
    float* out = (float*)d_out;

    // Workspace layout (~128.3 MB):
    char* ws = (char*)d_ws;
    float* support1 = (float*)(ws);                     // [100000,128] 51.2 MB
    float* agg1     = (float*)(ws + (size_t)51200000);  // [100000,128] 51.2 MB
    float* support2 = (float*)(ws + (size_t)102400000); // [100000, 32] 12.8 MB
    float* agg2     = (float*)(ws + (size_t)115200000); // [100000, 32] 12.8 MB
    float* W1t      = (float*)(ws + (size_t)128000000); // [128,512]    256 KB
    float* W2t      = (float*)(ws + (size_t)128262144); // [ 32,128]     16 KB

    hipMemsetAsync(agg1, 0, (size_t)N_NODES * F_HID * sizeof(float), stream);
    hipMemsetAsync(agg2, 0, (size_t)N_NODES * F_OUT * sizeof(float), stream);

    // 0) one-time weight transposes (tiny)
    transposeW<F_IN,  F_HID><<<(F_IN * F_HID + 255) / 256, 256, 0, stream>>>(W1, W1t);
    transposeW<F_HID, F_OUT><<<(F_HID * F_OUT + 255) / 256, 256, 0, stream>>>(W2, W2t);

    const int mtiles = N_NODES / 16;         // 6250 exactly
    const int gemmBlocks = (mtiles + 3) / 4; // 4 waves (128 threads) per block

    // 1) support1 = x @ W1                   (WMMA f32, clause-able b64 loads)
    wmma_gemm_f32<F_IN, F_HID, false>
        <<<gemmBlocks, 128, 0, stream>>>(x, W1t, nullptr, support1, N_NODES);

    // 2) agg1 = A @ support1                 (atomic scatter, 32 lanes/edge)
    {
        const long long total = (long long)N_EDGES * (F_HID / 4);
        const int blocks = (int)((total + 255) / 256);
        spmm_scatter<F_HID><<<blocks, 256, 0, stream>>>(erow, ecol, ev, support1, agg1);
    }

    // 3+4) support2 = relu(agg1 + b1) @ W2   (bias+ReLU fused into A loads)
    wmma_gemm_f32<F_HID, F_OUT, true>
        <<<gemmBlocks, 128, 0, stream>>>(agg1, W2t, b1, support2, N_NODES);

    // 5) agg2 = A @ support2                 (atomic scatter, 8 lanes/edge)
    {
        const long long total = (long long)N_EDGES * (F_OUT / 4);
        const int blocks = (int)((total + 255) / 256);
        spmm_scatter<F_OUT><<<blocks, 256, 0, stream>>>(erow, ecol, ev, support2, agg2);
    }

    // 6) out = log_softmax(agg2 + b2), one wave32 per node
    bias_logsoftmax32<<<(N_NODES + 3) / 4, 128, 0, stream>>>(agg2, b2, out);
}